// GraphAutoencoder_45268955300493
// MI455X (gfx1250) — compile-verified
//
#include <hip/hip_runtime.h>

#define N_NODES 100000
#define N_EDGES 1600000

typedef __attribute__((ext_vector_type(2))) float v2f;
typedef __attribute__((ext_vector_type(8))) float v8f;

__device__ __forceinline__ void atomic_add_f32(float* p, float v) {
#if defined(__AMDGCN__)
  unsafeAtomicAdd(p, v);   // lowers to global_atomic_add_f32 on CDNA
#else
  atomicAdd(p, v);
#endif
}

// ---------------------------------------------------------------------------
// Dense GEMM  H = X @ W  using V_WMMA_F32_16X16X4_F32 (native fp32 WMMA).
// One wave32 computes a 16 x FOUT output tile; block = 4 waves = 64 rows.
// W is staged once per block into LDS, re-laid-out in (k-pair, n) order so a
// lane's B fragment {W[k,n], W[k+1,n]} is one ds_load_b64. Pair-row stride is
// padded by 32 floats (128B) so the two half-waves hit disjoint LDS banks.
//
// A (16x4 f32) lane layout: lane l holds row m=l&15; VGPR0 k=(l<16?0:2),
// VGPR1 k=(l<16?1:3) -> a = {X[m,kk], X[m,kk+1]}, kk = k0 + hi*2  (b64 load).
// C/D (16x16 f32): VGPR j holds M = j + hi*8, N = l&15.
// ---------------------------------------------------------------------------
template<int FIN, int FOUT>
__global__ __launch_bounds__(128)
void gemm_wmma_f32(const float* __restrict__ X, const float* __restrict__ W,
                   float* __restrict__ H, int n) {
  constexpr int PAIRS = FIN / 2;
  constexpr int LROW  = FOUT * 2 + 32;   // floats per k-pair row (+128B pad)
  __shared__ float Wl[PAIRS * LROW];

  // Cooperative staging of W (FIN*FOUT <= 4096 floats), all threads join.
  for (int i = threadIdx.x; i < FIN * FOUT; i += 128) {
    const int k  = i / FOUT;
    const int nn = i - k * FOUT;
    Wl[(k >> 1) * LROW + nn * 2 + (k & 1)] = W[i];
  }
  __syncthreads();

  const int lane = threadIdx.x & 31;
  const int wave = threadIdx.x >> 5;
  const int row0 = (blockIdx.x * 4 + wave) * 16;
  if (row0 >= n) return;          // after the barrier: no divergent-exit hazard
  const int m  = lane & 15;
  const int hi = lane >> 4;

  const v8f vzero = {0.f, 0.f, 0.f, 0.f, 0.f, 0.f, 0.f, 0.f};
  v8f acc[FOUT / 16];
#pragma unroll
  for (int t = 0; t < FOUT / 16; ++t) acc[t] = vzero;

  const float* __restrict__ xrow = X + (long long)(row0 + m) * FIN;
#pragma unroll
  for (int k0 = 0; k0 < FIN; k0 += 4) {
    const int kk = k0 + hi * 2;
    const v2f a = *reinterpret_cast<const v2f*>(xrow + kk);          // b64
#pragma unroll
    for (int t = 0; t < FOUT / 16; ++t) {
      const v2f b = *reinterpret_cast<const v2f*>(
          &Wl[(kk >> 1) * LROW + (t * 16 + m) * 2]);                 // ds b64
      acc[t] = __builtin_amdgcn_wmma_f32_16x16x4_f32(
          /*neg_a=*/false, a, /*neg_b=*/false, b,
          /*c_mod=*/(short)0, acc[t], /*reuse_a=*/false, /*reuse_b=*/false);
    }
  }
#pragma unroll
  for (int t = 0; t < FOUT / 16; ++t)
#pragma unroll
    for (int j = 0; j < 8; ++j)
      H[(long long)(row0 + hi * 8 + j) * FOUT + t * 16 + m] = acc[t][j];
}

// ---------------------------------------------------------------------------
// Normalization: deg -> dinv = rsqrt(deg_edges + 1)
// ---------------------------------------------------------------------------
__global__ void zero_f32(float* __restrict__ p, int n) {
  int i = blockIdx.x * blockDim.x + threadIdx.x;
  if (i < n) p[i] = 0.0f;
}

__global__ void deg_accum(const long long* __restrict__ dst, float* __restrict__ deg, int e) {
  int i = blockIdx.x * blockDim.x + threadIdx.x;
  if (i < e) atomic_add_f32(&deg[dst[i]], 1.0f);
}

__global__ void deg_to_dinv(float* __restrict__ deg, int n) {
  int i = blockIdx.x * blockDim.x + threadIdx.x;
  if (i < n) {
    float d = deg[i] + 1.0f;  // self loop
    deg[i] = rsqrtf(fmaxf(d, 1.0f));
  }
}

// ---------------------------------------------------------------------------
// OUT[i,f] = H[i,f] * dinv[i]^2   (self-loop term; also initializes OUT)
// ---------------------------------------------------------------------------
__global__ void self_loop_init(const float* __restrict__ H, const float* __restrict__ dinv,
                               float* __restrict__ OUT, int total, int fshift) {
  int g = blockIdx.x * blockDim.x + threadIdx.x;
  if (g >= total) return;
  float di = dinv[g >> fshift];
  OUT[g] = H[g] * di * di;
}

// ---------------------------------------------------------------------------
// Edge scatter: OUT[dst,:] += H[src,:] * dinv[src]*dinv[dst]
// One thread = one 4-float chunk of one edge. cshift = log2(FOUT/4).
// H and OUT are ~25.6MB each: L2-resident on MI455X (192MB L2).
// ---------------------------------------------------------------------------
__global__ void scatter_edges(const float* __restrict__ H, const long long* __restrict__ src,
                              const long long* __restrict__ dst, const float* __restrict__ dinv,
                              float* __restrict__ OUT, int e, int cshift) {
  long long g = (long long)blockIdx.x * blockDim.x + threadIdx.x;
  long long ed = g >> cshift;
  if (ed >= e) return;
  int c = (int)(g & ((1 << cshift) - 1));
  long long s = src[ed], d = dst[ed];
  float nrm = dinv[s] * dinv[d];
  const int F = 4 << cshift;
  const float4 h4 = *reinterpret_cast<const float4*>(H + s * F + (c << 2));
  float* o = OUT + d * F + (c << 2);
  atomic_add_f32(o + 0, h4.x * nrm);
  atomic_add_f32(o + 1, h4.y * nrm);
  atomic_add_f32(o + 2, h4.z * nrm);
  atomic_add_f32(o + 3, h4.w * nrm);
}

// ---------------------------------------------------------------------------
// OUT += bias (broadcast over rows), optional ReLU. fmask = FOUT-1.
// ---------------------------------------------------------------------------
__global__ void bias_act(float* __restrict__ OUT, const float* __restrict__ b,
                         int total, int fmask, int relu) {
  int g = blockIdx.x * blockDim.x + threadIdx.x;
  if (g >= total) return;
  float v = OUT[g] + b[g & fmask];
  OUT[g] = relu ? fmaxf(v, 0.0f) : v;
}

// ---------------------------------------------------------------------------
// Host-side layer driver
// ---------------------------------------------------------------------------
static void run_gcn_layer(const float* X, const float* Wm, const float* bias,
                          float* OUT, float* H,
                          const long long* src, const long long* dst,
                          const float* dinv, int Fin, int Fout, bool relu,
                          hipStream_t s) {
  const int n = N_NODES;
  dim3 blk(128);
  dim3 grd(((n / 16) + 3) / 4);
  if (Fin == 64 && Fout == 64)
    gemm_wmma_f32<64, 64><<<grd, blk, 0, s>>>(X, Wm, H, n);
  else if (Fin == 64 && Fout == 32)
    gemm_wmma_f32<64, 32><<<grd, blk, 0, s>>>(X, Wm, H, n);
  else
    gemm_wmma_f32<32, 64><<<grd, blk, 0, s>>>(X, Wm, H, n);

  const int fshift = (Fout == 64) ? 6 : 5;
  const int cshift = fshift - 2;
  const int total = n * Fout;
  self_loop_init<<<(total + 255) / 256, 256, 0, s>>>(H, dinv, OUT, total, fshift);

  long long sthreads = (long long)N_EDGES << cshift;
  scatter_edges<<<(unsigned)((sthreads + 255) / 256), 256, 0, s>>>(
      H, src, dst, dinv, OUT, N_EDGES, cshift);

  bias_act<<<(total + 255) / 256, 256, 0, s>>>(OUT, bias, total, Fout - 1, relu ? 1 : 0);
}

extern "C" void kernel_launch(void* const* d_in, const int* in_sizes, int n_in,
                              void* d_out, int out_size, void* d_ws, size_t ws_size,
                              hipStream_t stream) {
  (void)in_sizes; (void)n_in; (void)out_size; (void)ws_size;

  const float* x = (const float*)d_in[0];
  const long long* ei = (const long long*)d_in[1];   // int64 edge_index [2, E]
  const long long* src = ei;
  const long long* dst = ei + N_EDGES;

  const float* W[6];
  const float* B[6];
  for (int i = 0; i < 6; ++i) {
    W[i] = (const float*)d_in[2 + 2 * i];
    B[i] = (const float*)d_in[3 + 2 * i];
  }

  float* out  = (float*)d_out;
  float* xrec = out;                                  // [N, 64]
  float* zbuf = out + (long long)N_NODES * 64;        // [N, 32]

  float* dinv = (float*)d_ws;                         // N floats (also deg scratch)
  float* H    = dinv + N_NODES;                       // N*64
  float* bufA = H    + (long long)N_NODES * 64;       // N*64
  float* bufB = bufA + (long long)N_NODES * 64;       // N*64

  // --- symmetric normalization ---
  zero_f32<<<(N_NODES + 255) / 256, 256, 0, stream>>>(dinv, N_NODES);
  deg_accum<<<(N_EDGES + 255) / 256, 256, 0, stream>>>(dst, dinv, N_EDGES);
  deg_to_dinv<<<(N_NODES + 255) / 256, 256, 0, stream>>>(dinv, N_NODES);

  // --- encoder ---
  run_gcn_layer(x,    W[0], B[0], bufA, H, src, dst, dinv, 64, 64, true,  stream);
  run_gcn_layer(bufA, W[1], B[1], bufB, H, src, dst, dinv, 64, 64, true,  stream);
  run_gcn_layer(bufB, W[2], B[2], zbuf, H, src, dst, dinv, 64, 32, false, stream);
  // --- decoder ---
  run_gcn_layer(zbuf, W[3], B[3], bufA, H, src, dst, dinv, 32, 64, true,  stream);
  run_gcn_layer(bufA, W[4], B[4], bufB, H, src, dst, dinv, 64, 64, true,  stream);
  run_gcn_layer(bufB, W[5], B[5], xrec, H, src, dst, dinv, 64, 64, false, stream);
}